// log_compression_10256381903615
// MI455X (gfx1250) — compile-verified
//
#include <hip/hip_runtime.h>
#include <hip/hip_bf16.h>

// Problem constants (from reference): x [B=128, C=64, L=8000] f32, out [B, C, L/2]
#define Bsz 128
#define Csz 64
#define Lsz 8000
#define EPS_LOG 1e-6f
#define EPS_BN  1e-5f

typedef float v2f __attribute__((ext_vector_type(2)));
typedef float v4f __attribute__((ext_vector_type(4)));
typedef float v8f __attribute__((ext_vector_type(8)));

// ---------------------------------------------------------------------------
// xor-permute within each 16-lane row via v_permlane16_b32 (no LDS, no waits).
// Lane-select nibble tables: sel_lo = lanes 0-7 of each row, sel_hi = lanes
// 8-15; lane i reads lane (i ^ mask) of its own row.
// ---------------------------------------------------------------------------
__device__ __forceinline__ float permlane16_xor(float v, unsigned sel_lo,
                                                unsigned sel_hi) {
    const unsigned u = __float_as_uint(v);
    const unsigned r =
        __builtin_amdgcn_permlane16(u, u, sel_lo, sel_hi, false, false);
    return __uint_as_float(r);
}

// ---------------------------------------------------------------------------
// Wave-level (sum, sumsq) reduction built on V_WMMA_F32_16X16X4_F32.
//
// B-matrix (4x16 f32, 2 VGPRs/lane):
//   lanes 0-15 : B[0][l] = b.x = s_l , B[1][l] = b.y = q_l
//   lanes 16-31: B[2][l-16] = b.x = s_l , B[3][l-16] = b.y = q_l
// A-matrix selector (16x4):
//   rows 0-7 : [1,0,1,0]  -> D[i][j] = s_j + s_{j+16}   (sum slots K=0,2)
//   rows 8-15: [0,1,0,1]  -> D[i][j] = q_j + q_{j+16}   (sumsq slots K=1,3)
// A lane layout: lane m (and m+16) hold A[m%16][{0,1}] ({2,3}) in a.x/a.y,
// so a.x = (lane%16 < 8), a.y = 1-a.x works for both halves.
// D VGPR0: lanes 0-15 = D[0][j] (pair-summed s), lanes 16-31 = D[8][j]
// (pair-summed q). Four permlane16 xor-steps inside each 16-lane row finish.
// Post: every lane 0-15 holds total S, every lane 16-31 holds total Q.
// EXEC must be all ones at the call site (callers use full, divergence-free
// waves).
// ---------------------------------------------------------------------------
__device__ __forceinline__ float wave_reduce_pair_wmma(float s, float q) {
    const unsigned lane = threadIdx.x & 31u;
    v2f a, b;
    b.x = s;
    b.y = q;
    const float sel = ((lane & 15u) < 8u) ? 1.0f : 0.0f;
    a.x = sel;
    a.y = 1.0f - sel;
    v8f c = {};
    c = __builtin_amdgcn_wmma_f32_16x16x4_f32(
        /*neg_a=*/false, a, /*neg_b=*/false, b,
        /*c_mod=*/(short)0, c, /*reuse_a=*/false, /*reuse_b=*/false);
    float r = c[0];
    r += permlane16_xor(r, 0x67452301u, 0xEFCDAB89u); // xor 1
    r += permlane16_xor(r, 0x54761032u, 0xDCFE98BAu); // xor 2
    r += permlane16_xor(r, 0x32107654u, 0xBA98FEDCu); // xor 4
    r += permlane16_xor(r, 0xFEDCBA98u, 0x76543210u); // xor 8
    return r; // lanes 0-15: S ; lanes 16-31: Q
}

__device__ __forceinline__ float logc(float v) {
    return __logf(fabsf(v) + EPS_LOG);
}

// ---------------------------------------------------------------------------
// Pass 1: one block per (b, c) row of 8000 floats (2000 b128 loads).
// Emits one (sum, sumsq) partial per block -> fully deterministic tree.
// ---------------------------------------------------------------------------
__global__ __launch_bounds__(256) void stats_partial_kernel(
    const v4f* __restrict__ x4, float2* __restrict__ partials) {
    const int b = blockIdx.x; // 0..127
    const int c = blockIdx.y; // 0..63
    const v4f* row = x4 + (size_t)(b * Csz + c) * (Lsz / 4);

    float s = 0.0f, q = 0.0f;
    for (int i = threadIdx.x; i < Lsz / 4; i += 256) {
        const v4f v = row[i];
        float y;
        y = logc(v.x); s += y; q = fmaf(y, y, q);
        y = logc(v.y); s += y; q = fmaf(y, y, q);
        y = logc(v.z); s += y; q = fmaf(y, y, q);
        y = logc(v.w); s += y; q = fmaf(y, y, q);
    }

    const float r = wave_reduce_pair_wmma(s, q);

    __shared__ float sArr[8];
    __shared__ float qArr[8];
    const int wave = threadIdx.x >> 5;
    const int lane = threadIdx.x & 31;
    if (lane == 0)  sArr[wave] = r;
    if (lane == 16) qArr[wave] = r;
    __syncthreads();
    if (threadIdx.x == 0) {
        float S = 0.0f, Q = 0.0f;
        for (int w = 0; w < 8; ++w) { S += sArr[w]; Q += qArr[w]; }
        partials[c * Bsz + b] = make_float2(S, Q);
    }
}

// ---------------------------------------------------------------------------
// Pass 2: one block per channel reduces its 128 partials, folds in
// gamma/beta into an affine (scale, shift) pair:
//   yn = y*scale + shift with scale = gamma*rsqrt(var+eps),
//   shift = beta - mean*scale.
// ---------------------------------------------------------------------------
__global__ __launch_bounds__(128) void stats_final_kernel(
    const float2* __restrict__ partials, const float* __restrict__ gamma,
    const float* __restrict__ beta, float2* __restrict__ params) {
    const int c = blockIdx.x;
    const float2 p = partials[c * Bsz + threadIdx.x];

    const float r = wave_reduce_pair_wmma(p.x, p.y);

    __shared__ float sArr[4];
    __shared__ float qArr[4];
    const int wave = threadIdx.x >> 5;
    const int lane = threadIdx.x & 31;
    if (lane == 0)  sArr[wave] = r;
    if (lane == 16) qArr[wave] = r;
    __syncthreads();
    if (threadIdx.x == 0) {
        float S = 0.0f, Q = 0.0f;
        for (int w = 0; w < 4; ++w) { S += sArr[w]; Q += qArr[w]; }
        const float invN  = 1.0f / (float)(Bsz * Lsz);
        const float mean  = S * invN;
        const float var   = fmaf(-mean, mean, Q * invN); // E[y^2] - mean^2
        const float scale = gamma[c] * rsqrtf(var + EPS_BN);
        const float shift = fmaf(-mean, scale, beta[c]);
        params[c] = make_float2(scale, shift);
    }
}

// ---------------------------------------------------------------------------
// Pass 3: fused log + affine-normalize + AvgPool1d(2). One v4f of output
// (4 pooled values) per thread from two v4f loads (8 inputs). Output is
// never reread -> nontemporal store so it doesn't pollute L2 (input is
// 262MB vs 192MB L2, partially resident at best).
// avg of two normalized values: 0.5*scale*(y0+y1) + shift.
// ---------------------------------------------------------------------------
__global__ __launch_bounds__(256) void fuse_pool_kernel(
    const v4f* __restrict__ x4, const float2* __restrict__ params,
    v4f* __restrict__ out4) {
    const int t = blockIdx.x * 256 + threadIdx.x; // 0 .. 8,191,999 (exact)
    const int j4  = t % 1000;        // v4f index within 4000-float out row
    const int row = t / 1000;        // b*Csz + c, 0..8191
    const int c   = row & (Csz - 1);

    const float2 ps = params[c];
    const float hs  = 0.5f * ps.x;

    const size_t ibase = (size_t)row * (Lsz / 4) + (size_t)j4 * 2;
    const v4f va = x4[ibase];
    const v4f vb = x4[ibase + 1];

    v4f o;
    o.x = fmaf(hs, logc(va.x) + logc(va.y), ps.y);
    o.y = fmaf(hs, logc(va.z) + logc(va.w), ps.y);
    o.z = fmaf(hs, logc(vb.x) + logc(vb.y), ps.y);
    o.w = fmaf(hs, logc(vb.z) + logc(vb.w), ps.y);

    __builtin_nontemporal_store(o, out4 + t);
}

// ---------------------------------------------------------------------------
extern "C" void kernel_launch(void* const* d_in, const int* in_sizes, int n_in,
                              void* d_out, int out_size, void* d_ws, size_t ws_size,
                              hipStream_t stream) {
    const v4f*   x4    = (const v4f*)d_in[0];   // [128,64,8000] f32
    const float* gamma = (const float*)d_in[1]; // [64]
    const float* beta  = (const float*)d_in[2]; // [64]
    v4f*         out4  = (v4f*)d_out;           // [128,64,4000] f32

    // Workspace layout (fully rewritten every call; no init needed):
    //   [0, 64KB)      float2 partials[64*128]
    //   [64KB, +512B)  float2 params[64]
    float2* partials = (float2*)d_ws;
    float2* params   = (float2*)((char*)d_ws + (size_t)Csz * Bsz * sizeof(float2));

    dim3 g1(Bsz, Csz);
    stats_partial_kernel<<<g1, 256, 0, stream>>>(x4, partials);
    stats_final_kernel<<<Csz, 128, 0, stream>>>(partials, gamma, beta, params);

    const int nOut4   = (Bsz * Csz * (Lsz / 2)) / 4; // 8,192,000
    const int nBlocks = nOut4 / 256;                 // 32,000 exact
    fuse_pool_kernel<<<nBlocks, 256, 0, stream>>>(x4, params, out4);
}